// GNN3_17471926960717
// MI455X (gfx1250) — compile-verified
//
#include <hip/hip_runtime.h>
#include <math.h>

#define N_DRUG 846
#define KNB    64
#define DIM    256
#define TWO_D  512
#define BN_EPS 1e-5f

typedef float v2f __attribute__((ext_vector_type(2)));
typedef float v8f __attribute__((ext_vector_type(8)));
typedef unsigned int v4u __attribute__((ext_vector_type(4)));
typedef int v4i __attribute__((ext_vector_type(4)));
typedef int v8i __attribute__((ext_vector_type(8)));

static __device__ __forceinline__ v8f v8f_zero() {
    v8f z = {0.f, 0.f, 0.f, 0.f, 0.f, 0.f, 0.f, 0.f};
    return z;
}

// D(16x16 f32) = A(16x4 f32) * B(4x16 f32) + C
static __device__ __forceinline__ v8f wmma_f32_4(v2f a, v2f b, v8f c) {
    return __builtin_amdgcn_wmma_f32_16x16x4_f32(false, a, false, b, (short)0, c,
                                                 false, false);
}

#define SW_STRIDE 272   // 256 + 16 pad; 2*272 mod 64 == 32 -> lane halves hit disjoint banks
#define SA_STRIDE 20    // 64 A-rows x 16 d; 20 mod 64 spreads rows over distinct banks

// ---------------------------------------------------------------------------
// TDM: DMA one W chunk [16 rows x 256 f32] global -> LDS.
// Hardware padding (pad_interval = 256 DWORDs, pad_amount = 16 DWORDs) lays the
// rows out at stride SW_STRIDE directly, matching the WMMA fragment reads.
// ---------------------------------------------------------------------------
static __device__ __forceinline__ void tdm_load_w_chunk(const float* gptr, float* lptr) {
    const unsigned long long ga = (unsigned long long)(uintptr_t)gptr;
    // LDS aperture: low 32 bits of the flat pointer == workgroup-relative LDS offset
    const unsigned int ldsoff = (unsigned int)(uintptr_t)lptr;

    v4u g0;
    g0.x = 1u;                                     // count=1 (valid user descriptor)
    g0.y = ldsoff;                                 // lds_addr (bytes)
    g0.z = (unsigned int)ga;                       // global_addr[31:0]
    g0.w = ((unsigned int)(ga >> 32) & 0x01FFFFFFu) | (2u << 30);  // addr[56:32] | type=2

    v8i g1;
    g1[0] = (int)((2u << 16)      // data_size = 4 bytes
                | (1u << 20)      // pad_enable
                | (7u << 22)      // pad_interval: 256 DWORDs (one 256-f32 row)
                | (15u << 25));   // pad_amount:   16 DWORDs  -> stride 272 f32
    g1[1] = (int)(256u << 16);    // tensor_dim0 = 256 (bits 79:48, low half)
    g1[2] = (int)(256u << 16);    // tensor_dim1 = 256 (bits 111:80, low half)
    g1[3] = (int)(256u << 16);    // tile_dim0 = 256 (bits 127:112)
    g1[4] = 16;                   // tile_dim1 = 16 rows
    g1[5] = 256;                  // tensor_dim0_stride = 256 elements
    g1[6] = 0;
    g1[7] = 0;

    v4i g2 = {0, 0, 0, 0};        // 2-D tile: groups 2/3 unused
    v4i g3 = {0, 0, 0, 0};
#if defined(__clang_major__) && (__clang_major__ >= 23)
    v8i g4 = {0, 0, 0, 0, 0, 0, 0, 0};
    __builtin_amdgcn_tensor_load_to_lds(g0, g1, g2, g3, g4, 0);
#else
    __builtin_amdgcn_tensor_load_to_lds(g0, g1, g2, g3, 0);
#endif
}

// ---------------------------------------------------------------------------
// Kernel 0: init — b2 row sums + zero BN accumulators
// ---------------------------------------------------------------------------
__global__ __launch_bounds__(256) void init_kernel(const float* __restrict__ b2,
                                                   float* __restrict__ b2row,
                                                   float* __restrict__ sums) {
    const int t = threadIdx.x;
    if (t < KNB) {
        float s = 0.f;
        for (int e = 0; e < DIM; ++e) s += b2[t * DIM + e];
        b2row[t] = s;
    }
    sums[t]       = 0.f;   // column sums
    sums[t + DIM] = 0.f;   // column sum-of-squares
}

// ---------------------------------------------------------------------------
// Kernel 1: per-node neighborhood aggregation.
//   h = relu(drug_rel @ W1[n] + b1)   (WMMA f32 16x16x4, h lives in VGPRs)
//   score = softmax_k( h @ rowsum_e(W2[n]) + rowsum_e(b2) )
//   drug_e[n] = [ sum_k score_k * ent_emb[n,k,:] , drug_emb[n] ]
// One 256-thread block (8 wave32) per node; W1 chunks arrive via TDM DMA,
// double-buffered against the WMMA loop.
// ---------------------------------------------------------------------------
__global__ __launch_bounds__(256) void gnn_node_kernel(
    const int* __restrict__ drug_name, const int* __restrict__ adj_tail,
    const int* __restrict__ adj_relation, const float* __restrict__ drug_table,
    const float* __restrict__ rela_table, const float* __restrict__ ent_table,
    const float* __restrict__ W1, const float* __restrict__ b1,
    const float* __restrict__ W2, const float* __restrict__ b2row,
    float* __restrict__ drug_e) {
    __shared__ __align__(16) float sW[2][16 * SW_STRIDE];   // W1 chunk (TDM dest)
    __shared__ __align__(16) float sA[2][KNB * SA_STRIDE];  // drug_rel chunk
    __shared__ __align__(16) float sV[DIM];                 // rowsum_e W2[n][d][:]
    __shared__ float sScore[KNB];
    __shared__ float sInvSum;

    const int n    = blockIdx.x;
    const int t    = threadIdx.x;
    const int wave = t >> 5;
    const int lane = t & 31;
    const int lo   = lane & 15;
    const int hi   = lane >> 4;
    const int m0   = (wave & 3) * 16;  // M-tile base row (neighbor index)
    const int eH   = wave >> 2;        // e-half: tiles [8*eH, 8*eH+8)

    const int    dn      = drug_name[n];
    const float* drugRow = drug_table + (size_t)dn * DIM;
    const float* w1base  = W1 + (size_t)n * DIM * DIM;

    // staging coordinates for the A (drug_rel) chunk
    const int    kRow    = t >> 2;       // neighbor row 0..63
    const int    i0      = (t & 3) * 4;  // d sub-offset 0,4,8,12
    const int    relIdx  = adj_relation[n * KNB + kRow];
    const float* relaRow = rela_table + (size_t)relIdx * DIM;

    if (t < KNB) sScore[t] = 0.0f;

    v8f acc[8];
#pragma unroll
    for (int i = 0; i < 8; ++i) acc[i] = v8f_zero();

    // ---- prologue: stage chunk 0 (A by ALU threads, W by TDM from wave 0) ----
    {
        const float4 dr = *reinterpret_cast<const float4*>(drugRow + i0);
        const float4 rl = *reinterpret_cast<const float4*>(relaRow + i0);
        float4 p;
        p.x = dr.x * rl.x; p.y = dr.y * rl.y;
        p.z = dr.z * rl.z; p.w = dr.w * rl.w;
        *reinterpret_cast<float4*>(&sA[0][kRow * SA_STRIDE + i0]) = p;
    }
    if (wave == 0) tdm_load_w_chunk(w1base, &sW[0][0]);

    // ---- GEMM1: h = drug_rel @ W1[n], reduction chunked by 16 ----
    for (int c = 0; c < 16; ++c) {
        const int cur = c & 1;
        if (wave == 0) __builtin_amdgcn_s_wait_tensorcnt(0);
        __syncthreads();   // sW[cur] (DMA) and sA[cur] (stores) visible to all

        if (c + 1 < 16) {  // stage next chunk while computing this one
            const int d0n = (c + 1) * 16;
            const float4 dr = *reinterpret_cast<const float4*>(drugRow + d0n + i0);
            const float4 rl = *reinterpret_cast<const float4*>(relaRow + d0n + i0);
            float4 p;
            p.x = dr.x * rl.x; p.y = dr.y * rl.y;
            p.z = dr.z * rl.z; p.w = dr.w * rl.w;
            *reinterpret_cast<float4*>(&sA[1 - cur][kRow * SA_STRIDE + i0]) = p;
            if (wave == 0)
                tdm_load_w_chunk(w1base + (size_t)d0n * DIM, &sW[1 - cur][0]);
        }

        const float* Wb = sW[cur];
        const float* Ab = sA[cur];
#pragma unroll
        for (int dl = 0; dl < 16; dl += 4) {
            const v2f a = *reinterpret_cast<const v2f*>(
                &Ab[(m0 + lo) * SA_STRIDE + dl + 2 * hi]);
#pragma unroll
            for (int et = 0; et < 8; ++et) {
                const int col = (eH * 8 + et) * 16 + lo;
                v2f b;
                b.x = Wb[(dl + 2 * hi) * SW_STRIDE + col];
                b.y = Wb[(dl + 2 * hi + 1) * SW_STRIDE + col];
                acc[et] = wmma_f32_4(a, b, acc[et]);
            }
        }
        __syncthreads();
    }

    // ---- rowsum_e(W2[n]) : thread t owns W2 row t (float4, WGP$-friendly) ----
    {
        const float* w2row = W2 + (size_t)n * DIM * DIM + (size_t)t * DIM;
        float s = 0.f;
#pragma unroll 4
        for (int j = 0; j < DIM; j += 4) {
            const float4 v = *reinterpret_cast<const float4*>(w2row + j);
            s += v.x + v.y + v.z + v.w;
        }
        sV[t] = s;
    }
    __syncthreads();

    // ---- per-row score partials: relu(h + b1) . w2sum, reduced via ds_add_f32 ----
#pragma unroll
    for (int j = 0; j < 8; ++j) {
        const int row = m0 + j + 8 * hi;
        float rp = 0.f;
#pragma unroll
        for (int et = 0; et < 8; ++et) {
            const int col = (eH * 8 + et) * 16 + lo;
            float v = acc[et][j] + b1[row * DIM + col];
            v = fmaxf(v, 0.f);
            rp += v * sV[col];
        }
        atomicAdd(&sScore[row], rp);
    }
    __syncthreads();

    // ---- softmax over the 64 neighbors ----
    if (t == 0) {
        float mx = -3.4e38f;
        for (int k = 0; k < KNB; ++k) mx = fmaxf(mx, sScore[k] + b2row[k]);
        float se = 0.f;
        for (int k = 0; k < KNB; ++k) {
            const float e = expf(sScore[k] + b2row[k] - mx);
            sScore[k] = e;
            se += e;
        }
        sInvSum = 1.f / se;
    }
    __syncthreads();
    if (t < KNB) sScore[t] *= sInvSum;
    __syncthreads();

    // ---- weighted neighbor aggregation + concat ----
    {
        float a2 = 0.f;
        const int* tails = adj_tail + n * KNB;
        for (int k = 0; k < KNB; ++k)
            a2 = fmaf(sScore[k], ent_table[(size_t)tails[k] * DIM + t], a2);
        drug_e[(size_t)n * TWO_D + t]       = a2;          // weighted
        drug_e[(size_t)n * TWO_D + DIM + t] = drugRow[t];  // drug_emb
    }
}

// ---------------------------------------------------------------------------
// Kernel 2: x = relu(drug_e @ lin_w + lin_b), [846,512]@[512,256] via WMMA f32.
// 53 blocks x 16 rows; 8 waves, 2 e-tiles each.
// ---------------------------------------------------------------------------
#define SA2_STRIDE 516
__global__ __launch_bounds__(256) void lin_kernel(const float* __restrict__ drug_e,
                                                  const float* __restrict__ lin_w,
                                                  const float* __restrict__ lin_b,
                                                  float* __restrict__ x) {
    __shared__ __align__(16) float sA[16 * SA2_STRIDE];
    const int t    = threadIdx.x;
    const int wave = t >> 5;
    const int lane = t & 31;
    const int lo   = lane & 15;
    const int hi   = lane >> 4;
    const int m0   = blockIdx.x * 16;

    // stage 16 rows of drug_e (zero-padded past N_DRUG)
#pragma unroll
    for (int j = 0; j < 8; ++j) {
        const int f   = t + 256 * j;  // 2048 float4s = 16 rows * 128
        const int row = f >> 7;
        const int c4  = f & 127;
        const int g   = m0 + row;
        float4 v = {0.f, 0.f, 0.f, 0.f};
        if (g < N_DRUG)
            v = *reinterpret_cast<const float4*>(drug_e + (size_t)g * TWO_D + c4 * 4);
        *reinterpret_cast<float4*>(&sA[row * SA2_STRIDE + c4 * 4]) = v;
    }
    __syncthreads();

    v8f acc0 = v8f_zero(), acc1 = v8f_zero();
    const int e0   = wave * 2;  // this wave's first e-tile
    const int col0 = e0 * 16 + lo;
#pragma unroll 4
    for (int d0 = 0; d0 < TWO_D; d0 += 4) {
        const v2f a = *reinterpret_cast<const v2f*>(&sA[lo * SA2_STRIDE + d0 + 2 * hi]);
        const int rb = (d0 + 2 * hi) * DIM;  // lin_w row base (L2 resident)
        v2f b0, b1v;
        b0.x  = lin_w[rb + col0];
        b0.y  = lin_w[rb + DIM + col0];
        b1v.x = lin_w[rb + col0 + 16];
        b1v.y = lin_w[rb + DIM + col0 + 16];
        acc0 = wmma_f32_4(a, b0, acc0);
        acc1 = wmma_f32_4(a, b1v, acc1);
    }

#pragma unroll
    for (int j = 0; j < 8; ++j) {
        const int row = m0 + j + 8 * hi;
        if (row < N_DRUG) {
            x[row * DIM + col0]      = fmaxf(acc0[j] + lin_b[col0], 0.f);
            x[row * DIM + col0 + 16] = fmaxf(acc1[j] + lin_b[col0 + 16], 0.f);
        }
    }
}

// ---------------------------------------------------------------------------
// Kernel 3a: column sums / sum-of-squares for BatchNorm (global f32 atomics)
// ---------------------------------------------------------------------------
__global__ __launch_bounds__(256) void bn_sum_kernel(const float* __restrict__ x,
                                                     float* __restrict__ sums) {
    const int t = threadIdx.x;
    float s = 0.f, s2 = 0.f;
    for (int r = blockIdx.x; r < N_DRUG; r += gridDim.x) {
        const float v = x[r * DIM + t];
        s += v;
        s2 += v * v;
    }
    atomicAdd(&sums[t], s);
    atomicAdd(&sums[t + DIM], s2);
}

// ---------------------------------------------------------------------------
// Kernel 3b: normalize (training-mode BN, biased variance)
// ---------------------------------------------------------------------------
__global__ __launch_bounds__(256) void bn_norm_kernel(
    const float* __restrict__ x, const float* __restrict__ sums,
    const float* __restrict__ bn_w, const float* __restrict__ bn_b,
    float* __restrict__ out) {
    const int t = threadIdx.x;
    const int r = blockIdx.x;
    const float inv  = 1.0f / (float)N_DRUG;
    const float mean = sums[t] * inv;
    const float var  = sums[t + DIM] * inv - mean * mean;
    const float v    = x[r * DIM + t];
    out[r * DIM + t] = (v - mean) * rsqrtf(var + BN_EPS) * bn_w[t] + bn_b[t];
}

// ---------------------------------------------------------------------------
extern "C" void kernel_launch(void* const* d_in, const int* in_sizes, int n_in,
                              void* d_out, int out_size, void* d_ws, size_t ws_size,
                              hipStream_t stream) {
    const int*   drug_name    = (const int*)d_in[0];
    const int*   adj_tail     = (const int*)d_in[1];
    const int*   adj_relation = (const int*)d_in[2];
    const float* drug_table   = (const float*)d_in[3];
    const float* rela_table   = (const float*)d_in[4];
    const float* ent_table    = (const float*)d_in[5];
    const float* W1           = (const float*)d_in[6];
    const float* b1           = (const float*)d_in[7];
    const float* W2           = (const float*)d_in[8];
    const float* b2           = (const float*)d_in[9];
    const float* lin_w        = (const float*)d_in[10];
    const float* lin_b        = (const float*)d_in[11];
    const float* bn_w         = (const float*)d_in[12];
    const float* bn_b         = (const float*)d_in[13];
    float*       out          = (float*)d_out;

    float* ws     = (float*)d_ws;
    float* drug_e = ws;                           // [846, 512]
    float* xbuf   = ws + (size_t)N_DRUG * TWO_D;  // [846, 256]
    float* sums   = xbuf + (size_t)N_DRUG * DIM;  // [512]
    float* b2row  = sums + 2 * DIM;               // [64]

    init_kernel<<<1, 256, 0, stream>>>(b2, b2row, sums);
    gnn_node_kernel<<<N_DRUG, 256, 0, stream>>>(drug_name, adj_tail, adj_relation,
                                                drug_table, rela_table, ent_table,
                                                W1, b1, W2, b2row, drug_e);
    lin_kernel<<<(N_DRUG + 15) / 16, 256, 0, stream>>>(drug_e, lin_w, lin_b, xbuf);
    bn_sum_kernel<<<32, 256, 0, stream>>>(xbuf, sums);
    bn_norm_kernel<<<846, 256, 0, stream>>>(xbuf, sums, bn_w, bn_b, out);
}